// AnomalyDAEBase_21887153340770
// MI455X (gfx1250) — compile-verified
//
#include <hip/hip_runtime.h>
#include <hip/hip_bf16.h>

#define NN   10000   // num nodes (625 * 16)
#define NNP  10016   // padded to 313 * 32 for the 32x32-tile decoder
#define IND  256     // in_dim
#define EMBD 64
#define HIDD 64
#define NE   320000  // edges (plus NN self loops)
#define NEG_SLOPE 0.2f

typedef float v2f __attribute__((ext_vector_type(2)));
typedef float v8f __attribute__((ext_vector_type(8)));

// ---------------- WMMA helpers (V_WMMA_F32_16X16X4_F32) ----------------

__device__ __forceinline__ v8f wmma4(v2f a, v2f b, v8f c) {
  // 8 args: (neg_a, A, neg_b, B, c_mod, C, reuse_a, reuse_b)
  return __builtin_amdgcn_wmma_f32_16x16x4_f32(false, a, false, b, (short)0, c,
                                               false, false);
}

// A-frag / Bt-frag from a row-major matrix X[. , ld]:
// lane l holds X[r0 + l%16][k + 2*(l/16) + {0,1}]  (one b64 load)
__device__ __forceinline__ v2f frag_rm(const float* __restrict__ X, int ld,
                                       int r0, int k, int lane) {
  return *(const v2f*)(X + (size_t)(r0 + (lane & 15)) * ld + k + 2 * (lane >> 4));
}

// B-frag from K-major B[k][n] (ld = row length):
// lane l holds B[k + 2*(l/16) + {0,1}][n0 + l%16]  (two strided b32 loads)
__device__ __forceinline__ v2f frag_km(const float* __restrict__ X, int ld,
                                       int n0, int k, int lane) {
  const float* p = X + (size_t)(k + 2 * (lane >> 4)) * ld + n0 + (lane & 15);
  v2f r;
  r.x = p[0];
  r.y = p[ld];
  return r;
}

// ordered-uint encoding for float atomic max
__device__ __forceinline__ unsigned f2ord(float f) {
  unsigned u = __float_as_uint(f);
  return (u >> 31) ? ~u : (u | 0x80000000u);
}
__device__ __forceinline__ float ord2f(unsigned k) {
  return __uint_as_float((k >> 31) ? (k ^ 0x80000000u) : ~k);
}

// ---------------- kernels ----------------

// h = relu(x @ W_stru^T + b_stru)   [10000,256]x[64,256]^T -> [10000,64]
__global__ void k_dense_stru(const float* __restrict__ x,
                             const float* __restrict__ W,
                             const float* __restrict__ b,
                             float* __restrict__ h) {
  int wave = (blockIdx.x * blockDim.x + threadIdx.x) >> 5;
  int lane = threadIdx.x & 31;
  if (wave >= 625 * 4) return;
  int m0 = (wave >> 2) * 16, n0 = (wave & 3) * 16;
  v8f c = {};
  for (int k = 0; k < IND; k += 4)
    c = wmma4(frag_rm(x, IND, m0, k, lane), frag_rm(W, IND, n0, k, lane), c);
  int col = n0 + (lane & 15);
  int rbase = m0 + ((lane >> 4) << 3);
  float bias = b[col];
#pragma unroll
  for (int i = 0; i < 8; ++i)
    h[(size_t)(rbase + i) * EMBD + col] = fmaxf(c[i] + bias, 0.f);
}

// g = h @ W_gat^T   [10000,64]x[64,64]^T -> [10000,64]
__global__ void k_gat_lin(const float* __restrict__ h,
                          const float* __restrict__ W,
                          float* __restrict__ g) {
  int wave = (blockIdx.x * blockDim.x + threadIdx.x) >> 5;
  int lane = threadIdx.x & 31;
  if (wave >= 625 * 4) return;
  int m0 = (wave >> 2) * 16, n0 = (wave & 3) * 16;
  v8f c = {};
  for (int k = 0; k < EMBD; k += 4)
    c = wmma4(frag_rm(h, EMBD, m0, k, lane), frag_rm(W, EMBD, n0, k, lane), c);
  int col = n0 + (lane & 15);
  int rbase = m0 + ((lane >> 4) << 3);
#pragma unroll
  for (int i = 0; i < 8; ++i)
    g[(size_t)(rbase + i) * HIDD + col] = c[i];
}

// a_s[n] = g[n].att_src ; a_d[n] = g[n].att_dst
__global__ void k_att_dots(const float* __restrict__ g,
                           const float* __restrict__ att_s,
                           const float* __restrict__ att_d,
                           float* __restrict__ a_s, float* __restrict__ a_d) {
  int n = blockIdx.x * blockDim.x + threadIdx.x;
  if (n >= NN) return;
  float s = 0.f, d = 0.f;
#pragma unroll 8
  for (int k = 0; k < HIDD; ++k) {
    float v = g[(size_t)n * HIDD + k];
    s += v * att_s[k];
    d += v * att_d[k];
  }
  a_s[n] = s;
  a_d[n] = d;
}

// init: mkey = ord(-inf), denom = 0, emb = b_gat broadcast (pad rows = 0),
// D (a1 split-K accumulator) = 0
__global__ void k_init(unsigned* __restrict__ mkey, float* __restrict__ denom,
                       float* __restrict__ emb, const float* __restrict__ b_gat,
                       float* __restrict__ D) {
  int i = blockIdx.x * blockDim.x + threadIdx.x;
  if (i < NN) { mkey[i] = 0u; denom[i] = 0.f; }
  if (i < IND * EMBD) D[i] = 0.f;
  if (i < NNP * HIDD) emb[i] = (i < NN * HIDD) ? b_gat[i & (HIDD - 1)] : 0.f;
}

__device__ __forceinline__ void edge_ids(const int* __restrict__ ei, int i,
                                         int& src, int& dst) {
  if (i < NE) { src = ei[i]; dst = ei[NE + i]; }
  else        { src = i - NE; dst = i - NE; }  // self loop
}

// pass 1: segment max of leaky_relu(a_s[src]+a_d[dst]) per dst
__global__ void k_edge_max(const int* __restrict__ ei,
                           const float* __restrict__ a_s,
                           const float* __restrict__ a_d,
                           unsigned* __restrict__ mkey) {
  int i = blockIdx.x * blockDim.x + threadIdx.x;
  if (i >= NE + NN) return;
  int src, dst;
  edge_ids(ei, i, src, dst);
  float e = a_s[src] + a_d[dst];
  e = e > 0.f ? e : NEG_SLOPE * e;
  atomicMax(mkey + dst, f2ord(e));
}

// pass 2: p = exp(e - m[dst]); denom[dst] += p
__global__ void k_edge_exp(const int* __restrict__ ei,
                           const float* __restrict__ a_s,
                           const float* __restrict__ a_d,
                           const unsigned* __restrict__ mkey,
                           float* __restrict__ p, float* __restrict__ denom) {
  int i = blockIdx.x * blockDim.x + threadIdx.x;
  if (i >= NE + NN) return;
  int src, dst;
  edge_ids(ei, i, src, dst);
  float e = a_s[src] + a_d[dst];
  e = e > 0.f ? e : NEG_SLOPE * e;
  float pv = __expf(e - ord2f(mkey[dst]));
  p[i] = pv;
  atomicAdd(denom + dst, pv);
}

// pass 3: emb[dst] += (p/denom[dst]) * g[src]   (one wave per edge, 2 feat/lane)
__global__ void k_edge_agg(const int* __restrict__ ei,
                           const float* __restrict__ p,
                           const float* __restrict__ denom,
                           const float* __restrict__ g,
                           float* __restrict__ emb) {
  int wid = (blockIdx.x * blockDim.x + threadIdx.x) >> 5;
  int lane = threadIdx.x & 31;
  if (wid >= NE + NN) return;
  int src, dst;
  edge_ids(ei, wid, src, dst);
  float alpha = p[wid] / denom[dst];
  v2f gv = *(const v2f*)(g + (size_t)src * HIDD + 2 * lane);
  atomicAdd(emb + (size_t)dst * HIDD + 2 * lane, alpha * gv.x);
  atomicAdd(emb + (size_t)dst * HIDD + 2 * lane + 1, alpha * gv.y);
}

// D += partial of (W_a1 @ x): A=W_a1 [64,10000] rm, B=x [10000,256] k-major.
// split-K (25 chunks of 400) for parallelism; atomicAdd into D [64,256].
__global__ void k_attr1(const float* __restrict__ Wa1,
                        const float* __restrict__ x, float* __restrict__ D) {
  int wave = (blockIdx.x * blockDim.x + threadIdx.x) >> 5;
  int lane = threadIdx.x & 31;
  if (wave >= 64 * 25) return;
  int tile = wave / 25, kc = wave % 25;
  int m0 = (tile >> 4) * 16, n0 = (tile & 15) * 16;
  v8f c = {};
  int k0 = kc * 400;
  for (int k = k0; k < k0 + 400; k += 4)
    c = wmma4(frag_rm(Wa1, NN, m0, k, lane), frag_km(x, IND, n0, k, lane), c);
  int col = n0 + (lane & 15);
  int rbase = m0 + ((lane >> 4) << 3);
#pragma unroll
  for (int i = 0; i < 8; ++i)
    atomicAdd(D + (size_t)(rbase + i) * IND + col, c[i]);
}

// a1T[e][i] = relu(D[e][i] + b_a1[e])
__global__ void k_attr1_relu(float* __restrict__ D, const float* __restrict__ b) {
  int i = blockIdx.x * blockDim.x + threadIdx.x;
  if (i >= EMBD * IND) return;
  D[i] = fmaxf(D[i] + b[i / IND], 0.f);
}

// a2T = W_a2 @ a1T + b_a2 : A=W_a2 [64,64] rm, B=a1T [64,256] k-major -> [64,256]
__global__ void k_attr2(const float* __restrict__ Wa2,
                        const float* __restrict__ a1T,
                        const float* __restrict__ b, float* __restrict__ a2T) {
  int wave = (blockIdx.x * blockDim.x + threadIdx.x) >> 5;
  int lane = threadIdx.x & 31;
  if (wave >= 4 * 16) return;
  int m0 = (wave >> 4) * 16, n0 = (wave & 15) * 16;
  v8f c = {};
  for (int k = 0; k < EMBD; k += 4)
    c = wmma4(frag_rm(Wa2, EMBD, m0, k, lane), frag_km(a1T, IND, n0, k, lane), c);
  int col = n0 + (lane & 15);
  int rbase = m0 + ((lane >> 4) << 3);
#pragma unroll
  for (int i = 0; i < 8; ++i)
    a2T[(size_t)(rbase + i) * IND + col] = c[i] + b[rbase + i];
}

// x_ = emb @ a2T : A=emb [10000,64] rm, B=a2T [64,256] k-major -> [10000,256]
__global__ void k_xhat(const float* __restrict__ emb,
                       const float* __restrict__ a2T, float* __restrict__ out) {
  int wave = (blockIdx.x * blockDim.x + threadIdx.x) >> 5;
  int lane = threadIdx.x & 31;
  if (wave >= 625 * 16) return;
  int m0 = (wave >> 4) * 16, n0 = (wave & 15) * 16;
  v8f c = {};
  for (int k = 0; k < HIDD; k += 4)
    c = wmma4(frag_rm(emb, HIDD, m0, k, lane), frag_km(a2T, IND, n0, k, lane), c);
  int col = n0 + (lane & 15);
  int rbase = m0 + ((lane >> 4) << 3);
#pragma unroll
  for (int i = 0; i < 8; ++i)
    out[(size_t)(rbase + i) * IND + col] = c[i];
}

// sigmoid-store of one 16x16 C tile with bounds guard (pad region dropped)
__device__ __forceinline__ void store_sig(float* __restrict__ s, v8f c, int m0,
                                          int n0, int lane) {
  int col = n0 + (lane & 15);
  if (col >= NN) return;
  int rbase = m0 + ((lane >> 4) << 3);
#pragma unroll
  for (int i = 0; i < 8; ++i) {
    int row = rbase + i;
    if (row < NN)
      s[(size_t)row * NN + col] = 1.f / (1.f + __expf(-c[i]));
  }
}

// s_ = sigmoid(emb @ emb^T) -> [10000,10000]  (dominant kernel)
// 32x32 output tile per wave: 4 accumulators, 2 A-frags + 2 B-frags per k-step
// -> 4 WMMAs per 4 b64 loads (2x the flops/load of a 16x16 tile).
__global__ void k_struct_dec(const float* __restrict__ emb,
                             float* __restrict__ s) {
  int wave = (blockIdx.x * blockDim.x + threadIdx.x) >> 5;
  int lane = threadIdx.x & 31;
  if (wave >= 313 * 313) return;
  int m0 = (wave / 313) * 32, n0 = (wave % 313) * 32;
  v8f c00 = {}, c01 = {}, c10 = {}, c11 = {};
#pragma unroll
  for (int k = 0; k < HIDD; k += 4) {
    v2f a0 = frag_rm(emb, HIDD, m0, k, lane);
    v2f a1 = frag_rm(emb, HIDD, m0 + 16, k, lane);
    v2f b0 = frag_rm(emb, HIDD, n0, k, lane);
    v2f b1 = frag_rm(emb, HIDD, n0 + 16, k, lane);
    c00 = wmma4(a0, b0, c00);
    c01 = wmma4(a0, b1, c01);
    c10 = wmma4(a1, b0, c10);
    c11 = wmma4(a1, b1, c11);
  }
  store_sig(s, c00, m0, n0, lane);
  store_sig(s, c01, m0, n0 + 16, lane);
  store_sig(s, c10, m0 + 16, n0, lane);
  store_sig(s, c11, m0 + 16, n0 + 16, lane);
}

// ---------------- launch ----------------

extern "C" void kernel_launch(void* const* d_in, const int* in_sizes, int n_in,
                              void* d_out, int out_size, void* d_ws,
                              size_t ws_size, hipStream_t stream) {
  const float* x      = (const float*)d_in[0];
  const int*   ei     = (const int*)d_in[1];
  const float* W_stru = (const float*)d_in[3];
  const float* b_stru = (const float*)d_in[4];
  const float* W_gat  = (const float*)d_in[5];
  const float* att_s  = (const float*)d_in[6];
  const float* att_d  = (const float*)d_in[7];
  const float* b_gat  = (const float*)d_in[8];
  const float* W_a1   = (const float*)d_in[9];
  const float* b_a1   = (const float*)d_in[10];
  const float* W_a2   = (const float*)d_in[11];
  const float* b_a2   = (const float*)d_in[12];

  float* ws = (float*)d_ws;
  float*    h     = ws;                 // 640000
  float*    g     = h + 640000;         // 640000
  float*    a_s   = g + 640000;         // 10000
  float*    a_d   = a_s + 10000;        // 10000
  unsigned* mkey  = (unsigned*)(a_d + 10000); // 10000
  float*    denom = (float*)(mkey + 10000);   // 10000
  float*    p     = denom + 10000;      // 330000
  float*    emb   = p + 330000;         // NNP*64 = 641024 (row-padded)
  float*    a1T   = emb + (size_t)NNP * HIDD; // 16384 (accum D, then relu'd)
  float*    a2T   = a1T + 16384;        // 16384

  float* out_x = (float*)d_out;                 // [10000,256]
  float* out_s = out_x + (size_t)NN * IND;      // [10000,10000]

  const int B = 256;
  // independent of edge kernels: init accumulators (covers emb padding too)
  k_init<<<(NNP * HIDD + B - 1) / B, B, 0, stream>>>(mkey, denom, emb, b_gat,
                                                     a1T);

  // structure encoder
  k_dense_stru<<<313, B, 0, stream>>>(x, W_stru, b_stru, h);
  k_gat_lin<<<313, B, 0, stream>>>(h, W_gat, g);
  k_att_dots<<<(NN + B - 1) / B, B, 0, stream>>>(g, att_s, att_d, a_s, a_d);

  // GAT softmax + aggregation
  const int EN = NE + NN;
  k_edge_max<<<(EN + B - 1) / B, B, 0, stream>>>(ei, a_s, a_d, mkey);
  k_edge_exp<<<(EN + B - 1) / B, B, 0, stream>>>(ei, a_s, a_d, mkey, p, denom);
  k_edge_agg<<<((size_t)EN * 32 + B - 1) / B, B, 0, stream>>>(ei, p, denom, g,
                                                              emb);

  // attribute path
  k_attr1<<<(64 * 25 * 32 + B - 1) / B, B, 0, stream>>>(W_a1, x, a1T);
  k_attr1_relu<<<(EMBD * IND + B - 1) / B, B, 0, stream>>>(a1T, b_a1);
  k_attr2<<<(64 * 32 + B - 1) / B, B, 0, stream>>>(W_a2, a1T, b_a2, a2T);

  // decoders
  k_xhat<<<(625 * 16 * 32 + B - 1) / B, B, 0, stream>>>(emb, a2T, out_x);
  k_struct_dec<<<(313 * 313 * 32 + B - 1) / B, B, 0, stream>>>(emb, out_s);
}